// RelationNetwork_36584531427735
// MI455X (gfx1250) — compile-verified
//
#include <hip/hip_runtime.h>

typedef __attribute__((ext_vector_type(2))) float v2f;
typedef __attribute__((ext_vector_type(8))) float v8f;

#define B_    4
#define N_    512
#define M_    512
#define D_    128
#define H_    256
#define DOUT_ 128

// =====================================================================
// Kernel 1: input projections via fp32 WMMA (16x16x4).
//   proj 0:  Apre[b*M+m][h] = Y[b,m,:] @ W1[0:D ,:] + b1[h]
//   proj 1:  Bpre[b*N+n][h] = X[b,n,:] @ W1[D:2D,:]
// One 16x16 output tile per wave; K = D = 128 -> 32 WMMA steps.
// =====================================================================
__global__ __launch_bounds__(256) void rn_proj_wmma(
    const float* __restrict__ X, const float* __restrict__ Y,
    const float* __restrict__ W1, const float* __restrict__ b1,
    float* __restrict__ ws)
{
  const int lane = threadIdx.x & 31;
  const int wave = threadIdx.x >> 5;
  int tile = blockIdx.x * 8 + wave;        // 4096 tiles total
  const int proj = tile >> 11;             // 0 = Y-projection, 1 = X-projection
  tile &= 2047;
  const int row0 = (tile >> 4) << 4;       // 16-row tile into flattened (B*M) rows
  const int h0   = (tile & 15) << 4;       // 16-col tile into H

  const float* src = (proj == 0 ? Y : X) + (size_t)row0 * D_;
  const float* w1p = W1 + (size_t)proj * D_ * H_;
  float*       dst = ws + (size_t)proj * ((size_t)B_ * M_ * H_);

  const int l16 = lane & 15;               // row (A) / col (B,C,D) within tile
  const int hi2 = (lane >> 4) << 1;        // K sub-offset (0 or 2) for this half-wave

  v8f c = {};
  #pragma unroll 4
  for (int k = 0; k < D_; k += 4) {
    const int kk = k + hi2;
    // A fragment: A[row=l16][K=kk,kk+1]  (contiguous 8B -> global_load_b64)
    const float* ap = src + (size_t)l16 * D_ + kk;
    v2f a; a.x = ap[0]; a.y = ap[1];
    // B fragment: W1[K=kk,kk+1][h0+l16]  (stride-H column loads)
    v2f b; b.x = w1p[(size_t)kk       * H_ + h0 + l16];
           b.y = w1p[(size_t)(kk + 1) * H_ + h0 + l16];
    c = __builtin_amdgcn_wmma_f32_16x16x4_f32(false, a, false, b,
                                              (short)0, c, false, false);
  }

  const float bias = (proj == 0) ? b1[h0 + l16] : 0.0f;
  const int rbase = row0 + ((lane >> 4) << 3);   // C/D: VGPR j holds row 8*(lane/16)+j
  #pragma unroll
  for (int j = 0; j < 8; ++j)
    dst[(size_t)(rbase + j) * H_ + h0 + l16] = c[j] + bias;
}

// =====================================================================
// Kernel 2: fused relu-sum + output GEMM.
// Block = (b, 16-row n-tile), 256 threads (8 waves).
//   Phase A (VALU-bound core): thread t owns column h=t.
//     s[i] = sum_m relu(Apre[b,m,t] + Bpre[b,n0+i,t]),  i = 0..15
//     One coalesced 1KB/block load of Apre per m, reused 16x in registers.
//   Phase B: s tile -> padded LDS, then out[b,n0:+16,:] = s @ W2 + M*b2
//     via fp32 WMMA, one 16x16 tile per wave (DOUT/16 = 8 waves).
// =====================================================================
#define SM_PITCH 260   // 256 + 4: conflict-free ds_load_b64 A-fragments

__global__ __launch_bounds__(256) void rn_main_wmma(
    const float* __restrict__ Apre,   // [B*M, H]  (b1 already folded in)
    const float* __restrict__ Bpre,   // [B*N, H]
    const float* __restrict__ W2,     // [H, DOUT]
    const float* __restrict__ b2,     // [DOUT]
    float* __restrict__ out)          // [B, N, DOUT]
{
  __shared__ float sm[16][SM_PITCH];

  const int b  = blockIdx.y;
  const int n0 = blockIdx.x << 4;
  const int t  = threadIdx.x;          // h column, 0..255

  const float* Arow = Apre + (size_t)b * M_ * H_ + t;
  const float* Bx   = Bpre + ((size_t)b * N_ + n0) * H_ + t;

  float bn[16];
  #pragma unroll
  for (int i = 0; i < 16; ++i) bn[i] = Bx[(size_t)i * H_];

  float s[16];
  #pragma unroll
  for (int i = 0; i < 16; ++i) s[i] = 0.0f;

  #pragma unroll 8
  for (int m = 0; m < M_; ++m) {
    const float a = Arow[(size_t)m * H_];
    #pragma unroll
    for (int i = 0; i < 16; ++i)
      s[i] += fmaxf(a + bn[i], 0.0f);
  }

  #pragma unroll
  for (int i = 0; i < 16; ++i) sm[i][t] = s[i];
  __syncthreads();

  // ---- Phase B: (16 x H) @ (H x DOUT) with WMMA 16x16x4 ----
  const int lane = t & 31;
  const int wave = t >> 5;
  const int o0   = wave << 4;          // output column tile
  const int l16  = lane & 15;
  const int hi2  = (lane >> 4) << 1;

  v8f c = {};
  #pragma unroll 4
  for (int k = 0; k < H_; k += 4) {
    const int kk = k + hi2;
    // A fragment from LDS: s[row=l16][K=kk,kk+1] (8B-aligned, conflict-free)
    v2f a; a.x = sm[l16][kk]; a.y = sm[l16][kk + 1];
    // B fragment: W2[K=kk,kk+1][o0+l16]
    v2f bb; bb.x = W2[(size_t)kk       * DOUT_ + o0 + l16];
            bb.y = W2[(size_t)(kk + 1) * DOUT_ + o0 + l16];
    c = __builtin_amdgcn_wmma_f32_16x16x4_f32(false, a, false, bb,
                                              (short)0, c, false, false);
  }

  const float bias = (float)M_ * b2[o0 + l16];     // sum_m b2 = M * b2
  const int rbase = (lane >> 4) << 3;              // n_local base for this half-wave
  #pragma unroll
  for (int j = 0; j < 8; ++j)
    out[((size_t)b * N_ + n0 + rbase + j) * DOUT_ + o0 + l16] = c[j] + bias;
}

// =====================================================================
// Launch: kernel1 fills ws (Apre @ offset 0, Bpre @ offset B*M*H floats),
// kernel2 consumes it. Same stream -> implicit ordering. ws need: 4 MB.
// =====================================================================
extern "C" void kernel_launch(void* const* d_in, const int* in_sizes, int n_in,
                              void* d_out, int out_size, void* d_ws, size_t ws_size,
                              hipStream_t stream) {
  (void)in_sizes; (void)n_in; (void)out_size;
  const float* X  = (const float*)d_in[0];
  const float* Y  = (const float*)d_in[1];
  const float* W1 = (const float*)d_in[2];
  const float* b1 = (const float*)d_in[3];
  const float* W2 = (const float*)d_in[4];
  const float* b2 = (const float*)d_in[5];
  float* out = (float*)d_out;
  float* ws  = (float*)d_ws;

  const size_t need = 2ull * B_ * M_ * H_ * sizeof(float);   // 4 MB
  if (ws_size < need) return;

  // 4096 wmma tiles / 8 waves per block = 512 blocks
  rn_proj_wmma<<<512, 256, 0, stream>>>(X, Y, W1, b1, ws);

  const float* Apre = ws;
  const float* Bpre = ws + (size_t)B_ * M_ * H_;
  rn_main_wmma<<<dim3(N_ / 16, B_), 256, 0, stream>>>(Apre, Bpre, W2, b2, out);
}